// GCN_jhgoh_32658931319628
// MI455X (gfx1250) — compile-verified
//
#include <hip/hip_runtime.h>
#include <hip/hip_bf16.h>
#include <math.h>

typedef float v2f __attribute__((ext_vector_type(2)));
typedef float v8f __attribute__((ext_vector_type(8)));

#define FEATS_IN 128
#define F1 32
#define F2 64
#define NGRAPH 256
#define BN_EPS 1e-5f

static __device__ __forceinline__ void atomicAddF(float* p, float v) {
  __hip_atomic_fetch_add(p, v, __ATOMIC_RELAXED, __HIP_MEMORY_SCOPE_AGENT);
}

// ---------------- degree ----------------
__global__ void deg_init_kernel(float* deg, int n) {
  int i = blockIdx.x * blockDim.x + threadIdx.x;
  if (i < n) deg[i] = 1.0f;  // self loop
}

__global__ void deg_count_kernel(const long long* __restrict__ dst, float* deg, int e) {
  int i = blockIdx.x * blockDim.x + threadIdx.x;
  if (i < e) atomicAddF(&deg[dst[i]], 1.0f);
}

__global__ void deg_rsqrt_kernel(float* deg, int n) {
  int i = blockIdx.x * blockDim.x + threadIdx.x;
  if (i < n) deg[i] = rsqrtf(deg[i]);  // in-place -> deg_inv
}

// ---------------- fp32 WMMA GEMM: C[M,N] = A[M,K] @ B[K,N] ----------------
// One wave per 16x16 output tile, K consumed 4 at a time by V_WMMA_F32_16X16X4_F32.
// M must be a multiple of 16 (100000 = 6250*16), N a multiple of 16, K of 4.
__global__ void gemm_wmma_f32(const float* __restrict__ A, const float* __restrict__ B,
                              float* __restrict__ C, int M, int K, int N) {
  int wave = (int)((blockIdx.x * (size_t)blockDim.x + threadIdx.x) >> 5);
  int lane = threadIdx.x & 31;
  int ntiles = N >> 4;
  int tiles = (M >> 4) * ntiles;
  if (wave >= tiles) return;
  int m0 = (wave / ntiles) << 4;
  int n0 = (wave % ntiles) << 4;
  int l15 = lane & 15;          // A row within tile / B col within tile
  int kof = (lane >> 4) << 1;   // upper 16 lanes hold K+2,K+3

  v8f acc = {};
  const float* Arow = A + (size_t)(m0 + l15) * K;
  for (int k0 = 0; k0 < K; k0 += 4) {
    v2f a, b;
    a.x = Arow[k0 + kof];
    a.y = Arow[k0 + kof + 1];
    b.x = B[(size_t)(k0 + kof) * N + n0 + l15];
    b.y = B[(size_t)(k0 + kof + 1) * N + n0 + l15];
    acc = __builtin_amdgcn_wmma_f32_16x16x4_f32(false, a, false, b, (short)0, acc,
                                                false, false);
  }
  int rbase = m0 + ((lane >> 4) << 3);  // upper lanes hold rows M=i+8
  for (int i = 0; i < 8; ++i) {
    C[(size_t)(rbase + i) * N + n0 + l15] = acc[i];
  }
}

// ---------------- self-loop + bias (also initializes scatter target) ----------------
__global__ void selfloop_init_kernel(const float* __restrict__ h, const float* __restrict__ dinv,
                                     const float* __restrict__ bias, float* __restrict__ out,
                                     int n, int f) {
  long long i = blockIdx.x * (long long)blockDim.x + threadIdx.x;
  if (i < (long long)n * f) {
    int node = (int)(i / f);
    int feat = (int)(i % f);
    float di = dinv[node];
    out[i] = di * di * h[i] + bias[feat];
  }
}

// ---------------- edge scatter: one wave per edge, lane = feature ----------------
__global__ void scatter32_kernel(const long long* __restrict__ src, const long long* __restrict__ dst,
                                 const float* __restrict__ dinv, const float* __restrict__ h,
                                 float* __restrict__ out, int e) {
  int wid = (int)((blockIdx.x * (size_t)blockDim.x + threadIdx.x) >> 5);
  int lane = threadIdx.x & 31;
  if (wid >= e) return;
  long long s = src[wid];
  long long d = dst[wid];
  float norm = dinv[s] * dinv[d];
  float v = h[s * F1 + lane] * norm;
  atomicAddF(&out[d * F1 + lane], v);
}

__global__ void scatter64_kernel(const long long* __restrict__ src, const long long* __restrict__ dst,
                                 const float* __restrict__ dinv, const float* __restrict__ h,
                                 float* __restrict__ out, int e) {
  int wid = (int)((blockIdx.x * (size_t)blockDim.x + threadIdx.x) >> 5);
  int lane = threadIdx.x & 31;
  if (wid >= e) return;
  long long s = src[wid];
  long long d = dst[wid];
  float norm = dinv[s] * dinv[d];
  float v0 = h[s * F2 + lane] * norm;
  float v1 = h[s * F2 + lane + 32] * norm;
  atomicAddF(&out[d * F2 + lane], v0);
  atomicAddF(&out[d * F2 + lane + 32], v1);
}

// ---------------- pooling ----------------
__global__ void pool_zero_kernel(float* sums, float* cnt) {
  int i = blockIdx.x * blockDim.x + threadIdx.x;
  if (i < NGRAPH * F2) sums[i] = 0.0f;
  if (i < NGRAPH) cnt[i] = 0.0f;
}

__global__ void pool_kernel(const float* __restrict__ h, const long long* __restrict__ batch,
                            float* sums, float* cnt, int n) {
  int wid = (int)((blockIdx.x * (size_t)blockDim.x + threadIdx.x) >> 5);
  int lane = threadIdx.x & 31;
  if (wid >= n) return;
  long long g = batch[wid];
  atomicAddF(&sums[g * F2 + lane], h[(size_t)wid * F2 + lane]);
  atomicAddF(&sums[g * F2 + lane + 32], h[(size_t)wid * F2 + lane + 32]);
  if (lane == 0) atomicAddF(&cnt[g], 1.0f);
}

// ---------------- head: Linear->ReLU->BatchNorm->Linear->Softplus (G=256) ----------------
__global__ void head_kernel(const float* __restrict__ sums, const float* __restrict__ cnt,
                            const float* __restrict__ fcW1, const float* __restrict__ fcb1,
                            const float* __restrict__ gamma, const float* __restrict__ beta,
                            const float* __restrict__ fcW2, const float* __restrict__ fcb2,
                            float* __restrict__ out) {
  __shared__ float zsh[NGRAPH * 32];
  __shared__ float mu[32];
  __shared__ float ivar[32];
  int g = threadIdx.x;  // 256 threads, one per graph

  float inv_c = 1.0f / fmaxf(cnt[g], 1.0f);
  for (int j = 0; j < 32; ++j) {
    float acc = fcb1[j];
    for (int k = 0; k < F2; ++k) {
      acc += sums[g * F2 + k] * inv_c * fcW1[k * 32 + j];
    }
    zsh[g * 32 + j] = fmaxf(acc, 0.0f);  // ReLU
  }
  __syncthreads();

  if (g < 32) {
    float s = 0.0f, s2 = 0.0f;
    for (int i = 0; i < NGRAPH; ++i) {
      float v = zsh[i * 32 + g];
      s += v;
      s2 += v * v;
    }
    float m = s * (1.0f / NGRAPH);
    mu[g] = m;
    float var = s2 * (1.0f / NGRAPH) - m * m;  // population variance (ddof=0)
    ivar[g] = rsqrtf(var + BN_EPS);
  }
  __syncthreads();

  float acc = fcb2[0];
  for (int j = 0; j < 32; ++j) {
    float zh = (zsh[g * 32 + j] - mu[j]) * ivar[j] * gamma[j] + beta[j];
    acc += zh * fcW2[j];
  }
  // softplus with overflow guard
  out[g] = (acc > 20.0f) ? acc : log1pf(expf(acc));
}

static inline int cdiv(long long a, long long b) { return (int)((a + b - 1) / b); }

extern "C" void kernel_launch(void* const* d_in, const int* in_sizes, int n_in,
                              void* d_out, int out_size, void* d_ws, size_t ws_size,
                              hipStream_t stream) {
  const float* x         = (const float*)d_in[0];
  const long long* ei    = (const long long*)d_in[1];
  const long long* batch = (const long long*)d_in[2];
  const float* W1   = (const float*)d_in[3];
  const float* b1   = (const float*)d_in[4];
  const float* W2   = (const float*)d_in[5];
  const float* b2   = (const float*)d_in[6];
  const float* fcW1 = (const float*)d_in[7];
  const float* fcb1 = (const float*)d_in[8];
  const float* gam  = (const float*)d_in[9];
  const float* bet  = (const float*)d_in[10];
  const float* fcW2 = (const float*)d_in[11];
  const float* fcb2 = (const float*)d_in[12];

  const int N = in_sizes[0] / FEATS_IN;
  const int E = in_sizes[1] / 2;
  const long long* src = ei;
  const long long* dst = ei + E;

  // workspace layout (floats), 64-element aligned slabs
  float* ws = (float*)d_ws;
  size_t off = 0;
  float* deg  = ws + off; off += ((size_t)N + 63) & ~(size_t)63;   // -> deg_inv in place
  float* h1   = ws + off; off += (size_t)N * F1;
  float* o1   = ws + off; off += (size_t)N * F1;
  float* h2   = ws + off; off += (size_t)N * F2;
  float* o2   = ws + off; off += (size_t)N * F2;
  float* sums = ws + off; off += (size_t)NGRAPH * F2;
  float* cnt  = ws + off; off += (size_t)NGRAPH;
  (void)ws_size;

  const int B = 256;

  // degree with self loops -> rsqrt
  deg_init_kernel<<<cdiv(N, B), B, 0, stream>>>(deg, N);
  deg_count_kernel<<<cdiv(E, B), B, 0, stream>>>(dst, deg, E);
  deg_rsqrt_kernel<<<cdiv(N, B), B, 0, stream>>>(deg, N);

  // conv1: h1 = x @ W1 ; o1 = dinv^2*h1 + b1 ; o1[d] += norm*h1[s]
  {
    long long tiles = (long long)(N / 16) * (F1 / 16);
    gemm_wmma_f32<<<cdiv(tiles * 32, B), B, 0, stream>>>(x, W1, h1, N, FEATS_IN, F1);
  }
  selfloop_init_kernel<<<cdiv((long long)N * F1, B), B, 0, stream>>>(h1, deg, b1, o1, N, F1);
  scatter32_kernel<<<cdiv((long long)E * 32, B), B, 0, stream>>>(src, dst, deg, h1, o1, E);

  // conv2: h2 = o1 @ W2 ; o2 = dinv^2*h2 + b2 ; o2[d] += norm*h2[s]
  {
    long long tiles = (long long)(N / 16) * (F2 / 16);
    gemm_wmma_f32<<<cdiv(tiles * 32, B), B, 0, stream>>>(o1, W2, h2, N, F1, F2);
  }
  selfloop_init_kernel<<<cdiv((long long)N * F2, B), B, 0, stream>>>(h2, deg, b2, o2, N, F2);
  scatter64_kernel<<<cdiv((long long)E * 32, B), B, 0, stream>>>(src, dst, deg, h2, o2, E);

  // scatter-mean pooling
  pool_zero_kernel<<<cdiv(NGRAPH * F2, B), B, 0, stream>>>(sums, cnt);
  pool_kernel<<<cdiv((long long)N * 32, B), B, 0, stream>>>(o2, batch, sums, cnt, N);

  // MLP head, one block of 256 threads
  head_kernel<<<1, NGRAPH, 0, stream>>>(sums, cnt, fcW1, fcb1, gam, bet, fcW2, fcb2,
                                        (float*)d_out);
}